// RetinaNetHead_18511309046024
// MI455X (gfx1250) — compile-verified
//
#include <hip/hip_runtime.h>
#include <hip/hip_bf16.h>
#include <stdint.h>

#define BATCH       32
#define NLEV        5
#define ATOT        196416
#define KEEP_TOTAL  4576
#define POST        300
#define IMG_F       1024.0f
#define BBOX_CLIP_F 4.135166556742356f  /* log(1000/16) */

typedef __attribute__((ext_vector_type(2))) float v2f;
typedef __attribute__((ext_vector_type(8))) float v8f;

/* ---------------- level tables ---------------- */
__device__ __forceinline__ int lev_n(int l)   { int f = 128 >> l; return 9 * f * f; }
__device__ __forceinline__ int lev_off(int l) { const int o[5] = {0,147456,184320,193536,195840}; return o[l]; }
__device__ __forceinline__ int lev_k(int l)   { return (l < 4) ? 1000 : 576; }
__device__ __forceinline__ int keep_off(int l){ return l * 1000; }

/* order-preserving float -> uint key (ascending) */
__device__ __forceinline__ unsigned fkey(float f) {
  unsigned u = __float_as_uint(f);
  return (u & 0x80000000u) ? ~u : (u | 0x80000000u);
}

/* ------- CDNA5 async global->LDS staging (gfx1250 VGLOBAL async ops) ------- */
__device__ __forceinline__ void async_g2lds_b128(void* lds_dst, const void* g_src) {
  unsigned l = (unsigned)(uintptr_t)lds_dst;          /* low 32b of flat LDS addr = LDS offset */
  uint64_t g = (uint64_t)(uintptr_t)g_src;
  asm volatile("global_load_async_to_lds_b128 %0, %1, off" :: "v"(l), "v"(g) : "memory");
}
__device__ __forceinline__ void async_g2lds_b32(void* lds_dst, const void* g_src) {
  unsigned l = (unsigned)(uintptr_t)lds_dst;
  uint64_t g = (uint64_t)(uintptr_t)g_src;
  asm volatile("global_load_async_to_lds_b32 %0, %1, off" :: "v"(l), "v"(g) : "memory");
}
__device__ __forceinline__ void wait_async0() {
  asm volatile("s_wait_asynccnt 0" ::: "memory");
}

/* =====================================================================
 * Kernel 1: per-(batch,level) exact top-k via 4-pass radix select.
 * The 256-bin suffix scan each pass is computed by wave 0 with
 * V_WMMA_F32_16X16X4_F32:  suffix = H*U + (L*H)*J   (exact in f32,
 * counts < 2^24). grid = BATCH*NLEV blocks of 256 threads.
 * ===================================================================== */
__global__ __launch_bounds__(256) void topk_select(
    const float* __restrict__ logits,   /* BATCH x ATOT */
    int*         __restrict__ keep_idx) /* BATCH x KEEP_TOTAL */
{
  const int blk = blockIdx.x;
  const int b   = blk / NLEV;
  const int lev = blk % NLEV;
  const int n   = lev_n(lev);
  const int off = lev_off(lev);
  const int k   = lev_k(lev);
  const int t   = threadIdx.x;
  const float* __restrict__ src = logits + (size_t)b * ATOT + off;

  __shared__ unsigned hist[256];
  __shared__ unsigned suffix[257];
  __shared__ float    s_P[256];          /* intermediate P = L*H */
  __shared__ unsigned sh_prefix, sh_remaining;
  __shared__ unsigned cnt_gt, cnt_eq;

  if (t == 0) { sh_prefix = 0u; sh_remaining = (unsigned)k; }

  for (int pass = 0; pass < 4; ++pass) {
    const int shift = 24 - 8 * pass;
    hist[t] = 0u;
    __syncthreads();
    const unsigned pfx = sh_prefix;
    for (int i = t; i < n; i += 256) {
      if (i + 2048 < n) __builtin_prefetch(src + i + 2048, 0, 0);  /* global_prefetch_b8 */
      unsigned key = fkey(src[i]);
      bool sel = (pass == 0) || ((key >> (shift + 8)) == pfx);
      if (sel) atomicAdd(&hist[(key >> shift) & 0xFFu], 1u);
    }
    __syncthreads();

    /* ---- wave 0: suffix scan of hist via WMMA (EXEC all-1s in wave 0) ---- */
    if (t < 32) {
      const int half = t >> 4;        /* 0: lanes 0-15, 1: lanes 16-31 */
      const int q    = t & 15;        /* M-row for A operands, N-col for B operands */

      /* Step 1: P = L x H  with  L[m][k] = (k > m) */
      v8f accP = {};
#pragma unroll
      for (int k0 = 0; k0 < 16; k0 += 4) {
        const int kA0 = k0 + (half ? 2 : 0);   /* A/B VGPR0 K-index (ISA 16x4 f32 layout) */
        const int kA1 = k0 + (half ? 3 : 1);   /* A/B VGPR1 K-index */
        v2f a, bv;
        a.x  = (kA0 > q) ? 1.0f : 0.0f;        /* A = L, row m = q */
        a.y  = (kA1 > q) ? 1.0f : 0.0f;
        bv.x = (float)hist[16 * kA0 + q];      /* B = H, col n = q */
        bv.y = (float)hist[16 * kA1 + q];
        accP = __builtin_amdgcn_wmma_f32_16x16x4_f32(
            false, a, false, bv, (short)0, accP, false, false);
      }
#pragma unroll
      for (int v = 0; v < 8; ++v) {            /* D layout: M = v + 8*half, N = q */
        s_P[16 * (v + 8 * half) + q] = accP[v];
      }

      /* Step 2: R = H x U + P x J  with  U[k][n] = (k >= n), J = ones */
      v8f accR = {};
#pragma unroll
      for (int k0 = 0; k0 < 16; k0 += 4) {
        const int kA0 = k0 + (half ? 2 : 0);
        const int kA1 = k0 + (half ? 3 : 1);
        v2f a, bv;
        a.x  = (float)hist[16 * q + kA0];      /* A = H, row m = q */
        a.y  = (float)hist[16 * q + kA1];
        bv.x = (kA0 >= q) ? 1.0f : 0.0f;       /* B = U, col n = q */
        bv.y = (kA1 >= q) ? 1.0f : 0.0f;
        accR = __builtin_amdgcn_wmma_f32_16x16x4_f32(
            false, a, false, bv, (short)0, accR, false, false);
      }
#pragma unroll
      for (int k0 = 0; k0 < 16; k0 += 4) {
        const int kA0 = k0 + (half ? 2 : 0);
        const int kA1 = k0 + (half ? 3 : 1);
        v2f a, bv;
        a.x  = s_P[16 * q + kA0];              /* A = P, row m = q */
        a.y  = s_P[16 * q + kA1];
        bv.x = 1.0f;                           /* B = J */
        bv.y = 1.0f;
        accR = __builtin_amdgcn_wmma_f32_16x16x4_f32(
            false, a, false, bv, (short)0, accR, false, false);
      }
#pragma unroll
      for (int v = 0; v < 8; ++v) {
        suffix[16 * (v + 8 * half) + q] = (unsigned)(accR[v] + 0.5f);
      }
      if (t == 0) suffix[256] = 0u;
    }
    __syncthreads();

    if (t == 0) {
      unsigned rem = sh_remaining, d = 0u;
      for (int j = 255; j >= 0; --j) { if (suffix[j] >= rem) { d = (unsigned)j; break; } }
      unsigned gt = suffix[d + 1];
      sh_remaining = rem - gt;
      sh_prefix    = (pfx << 8) | d;
    }
    __syncthreads();
  }

  const unsigned T   = sh_prefix;               /* threshold key            */
  const unsigned rem = sh_remaining;            /* #equal-to-T to keep      */
  const unsigned gtN = (unsigned)k - rem;       /* #strictly-greater (exact)*/
  if (t == 0) { cnt_gt = 0u; cnt_eq = 0u; }
  __syncthreads();

  int* __restrict__ dst = keep_idx + (size_t)b * KEEP_TOTAL + keep_off(lev);
  for (int i = t; i < n; i += 256) {
    unsigned key = fkey(src[i]);
    if (key > T) {
      unsigned p = atomicAdd(&cnt_gt, 1u);
      dst[p] = off + i;
    } else if (key == T) {
      unsigned p = atomicAdd(&cnt_eq, 1u);
      if (p < rem) dst[gtN + p] = off + i;
    }
  }
}

/* =====================================================================
 * Kernel 2: gather + decode + clip + min-size filter for kept candidates.
 * one thread per (batch, keep-slot).
 * ===================================================================== */
__global__ __launch_bounds__(256) void gather_decode(
    const float* __restrict__ logits,      /* B x ATOT        */
    const float* __restrict__ deltas,      /* B x ATOT x 4    */
    const float* __restrict__ anchors,     /* ATOT x 4        */
    const int*   __restrict__ image_sizes, /* B x 2 (h, w)    */
    const int*   __restrict__ keep_idx,    /* B x KEEP_TOTAL  */
    float*       __restrict__ box_off,     /* B x K x 4 (lvl-offset, for IoU) */
    float*       __restrict__ box_raw,     /* B x K x 4 (clipped, for output) */
    float*       __restrict__ score,       /* B x K */
    float*       __restrict__ area)        /* B x K */
{
  int g = blockIdx.x * 256 + threadIdx.x;
  if (g >= BATCH * KEEP_TOTAL) return;
  const int b = g / KEEP_TOTAL;
  const int j = g % KEEP_TOTAL;
  const int lvl = j / 1000;                    /* keep slots are per-level blocks */
  const int i = keep_idx[g];

  const float* a = anchors + 4 * (size_t)i;
  const float* d = deltas + ((size_t)b * ATOT + (size_t)i) * 4;

  float aw = a[2] - a[0], ah = a[3] - a[1];
  float cx = a[0] + 0.5f * aw, cy = a[1] + 0.5f * ah;
  float dw = fminf(d[2], BBOX_CLIP_F), dh = fminf(d[3], BBOX_CLIP_F);
  float pcx = d[0] * aw + cx, pcy = d[1] * ah + cy;
  float pw = __expf(dw) * aw, ph = __expf(dh) * ah;

  float x1 = pcx - 0.5f * pw, y1 = pcy - 0.5f * ph;
  float x2 = pcx + 0.5f * pw, y2 = pcy + 0.5f * ph;

  float hImg = (float)image_sizes[2 * b + 0];
  float wImg = (float)image_sizes[2 * b + 1];
  x1 = fminf(fmaxf(x1, 0.f), wImg);  x2 = fminf(fmaxf(x2, 0.f), wImg);
  y1 = fminf(fmaxf(y1, 0.f), hImg);  y2 = fminf(fmaxf(y2, 0.f), hImg);

  float lg = logits[(size_t)b * ATOT + i];
  float s  = 1.f / (1.f + __expf(-lg));
  bool ok  = ((x2 - x1) >= 1.0f) && ((y2 - y1) >= 1.0f);
  s = ok ? s : -1.0f;

  float o = (float)lvl * (IMG_F + 1.0f);
  box_raw[4 * g + 0] = x1;      box_raw[4 * g + 1] = y1;
  box_raw[4 * g + 2] = x2;      box_raw[4 * g + 3] = y2;
  box_off[4 * g + 0] = x1 + o;  box_off[4 * g + 1] = y1 + o;
  box_off[4 * g + 2] = x2 + o;  box_off[4 * g + 3] = y2 + o;
  score[g] = s;
  area[g]  = (x2 - x1) * (y2 - y1);
}

/* =====================================================================
 * Kernel 3: greedy NMS, one block per batch image. Entire candidate set
 * staged into LDS with CDNA5 async global->LDS loads (~110 KB, needs the
 * 320 KB WGP LDS); 300 iterations of argmax + suppression out of LDS.
 * ===================================================================== */
__global__ __launch_bounds__(256) void nms_kernel(
    const float* __restrict__ box_off,
    const float* __restrict__ box_raw,
    const float* __restrict__ score,
    const float* __restrict__ area,
    float*       __restrict__ out_boxes,   /* B x POST x 4 */
    float*       __restrict__ out_scores)  /* B x POST     */
{
  __shared__ __align__(16) float s_box[KEEP_TOTAL * 4];
  __shared__ float s_sc[KEEP_TOTAL];
  __shared__ float s_ar[KEEP_TOTAL];
  __shared__ float red_s[256];
  __shared__ int   red_i[256];

  const int b = blockIdx.x;
  const int t = threadIdx.x;
  const float* gb = box_off + (size_t)b * KEEP_TOTAL * 4;
  const float* gs = score   + (size_t)b * KEEP_TOTAL;
  const float* ga = area    + (size_t)b * KEEP_TOTAL;

  /* ---- async fill of LDS working set ---- */
  for (int i = 4 * t; i < KEEP_TOTAL * 4; i += 4 * 256)
    async_g2lds_b128(&s_box[i], &gb[i]);
  for (int i = t; i < KEEP_TOTAL; i += 256) {
    async_g2lds_b32(&s_sc[i], &gs[i]);
    async_g2lds_b32(&s_ar[i], &ga[i]);
  }
  wait_async0();
  __syncthreads();

  for (int it = 0; it < POST; ++it) {
    /* argmax with first-occurrence tie-break (matches jnp.argmax) */
    float bs = -__builtin_inff();
    int   bi = KEEP_TOTAL;
    for (int j = t; j < KEEP_TOTAL; j += 256) {
      float s = s_sc[j];
      if (s > bs || (s == bs && j < bi)) { bs = s; bi = j; }
    }
    red_s[t] = bs; red_i[t] = bi;
    __syncthreads();
    for (int w = 128; w > 0; w >>= 1) {
      if (t < w) {
        float os = red_s[t + w]; int oi = red_i[t + w];
        if (os > red_s[t] || (os == red_s[t] && oi < red_i[t])) {
          red_s[t] = os; red_i[t] = oi;
        }
      }
      __syncthreads();
    }
    const int   win = red_i[0];
    const float ws  = red_s[0];
    __syncthreads();

    const float wx1 = s_box[4 * win + 0], wy1 = s_box[4 * win + 1];
    const float wx2 = s_box[4 * win + 2], wy2 = s_box[4 * win + 3];
    const float wa  = s_ar[win];

    if (t == 0) {
      bool valid = ws > 0.0f;
      const float* rb = box_raw + ((size_t)b * KEEP_TOTAL + win) * 4;
      float* ob = out_boxes + ((size_t)b * POST + it) * 4;
      ob[0] = valid ? rb[0] : 0.0f;
      ob[1] = valid ? rb[1] : 0.0f;
      ob[2] = valid ? rb[2] : 0.0f;
      ob[3] = valid ? rb[3] : 0.0f;
      out_scores[(size_t)b * POST + it] = valid ? ws : 0.0f;
    }

    /* suppress: iou > 0.5  <=>  inter > 0.5 * union (union >= 1e-6 > 0) */
    for (int j = t; j < KEEP_TOTAL; j += 256) {
      float xx1 = fmaxf(wx1, s_box[4 * j + 0]);
      float yy1 = fmaxf(wy1, s_box[4 * j + 1]);
      float xx2 = fminf(wx2, s_box[4 * j + 2]);
      float yy2 = fminf(wy2, s_box[4 * j + 3]);
      float inter = fmaxf(xx2 - xx1, 0.0f) * fmaxf(yy2 - yy1, 0.0f);
      float uni   = fmaxf(wa + s_ar[j] - inter, 1e-6f);
      if (inter > 0.5f * uni) s_sc[j] = -__builtin_inff();
    }
    __syncthreads();
  }
}

/* ===================================================================== */
extern "C" void kernel_launch(void* const* d_in, const int* in_sizes, int n_in,
                              void* d_out, int out_size, void* d_ws, size_t ws_size,
                              hipStream_t stream) {
  (void)in_sizes; (void)n_in; (void)out_size; (void)ws_size;
  const float* pred_class  = (const float*)d_in[0];   /* B x ATOT x 1 */
  const float* pred_deltas = (const float*)d_in[1];   /* B x ATOT x 4 */
  const float* anchors     = (const float*)d_in[2];   /* ATOT x 4     */
  const int*   image_sizes = (const int*)  d_in[3];   /* B x 2        */

  char* ws = (char*)d_ws;
  const size_t KEEP_BYTES = (size_t)BATCH * KEEP_TOTAL * sizeof(int);      /* 585,728   */
  const size_t BOX_BYTES  = (size_t)BATCH * KEEP_TOTAL * 4 * sizeof(float);/* 2,342,912 */
  const size_t SC_BYTES   = (size_t)BATCH * KEEP_TOTAL * sizeof(float);    /* 585,728   */

  int*   keep    = (int*)  (ws);
  float* box_off = (float*)(ws + KEEP_BYTES);
  float* box_raw = (float*)(ws + KEEP_BYTES + BOX_BYTES);
  float* sc      = (float*)(ws + KEEP_BYTES + 2 * BOX_BYTES);
  float* ar      = (float*)(ws + KEEP_BYTES + 2 * BOX_BYTES + SC_BYTES);

  topk_select<<<BATCH * NLEV, 256, 0, stream>>>(pred_class, keep);

  int nCand = BATCH * KEEP_TOTAL;
  gather_decode<<<(nCand + 255) / 256, 256, 0, stream>>>(
      pred_class, pred_deltas, anchors, image_sizes, keep,
      box_off, box_raw, sc, ar);

  float* ob = (float*)d_out;                       /* B x POST x 4 */
  float* os = ob + (size_t)BATCH * POST * 4;       /* B x POST     */
  nms_kernel<<<BATCH, 256, 0, stream>>>(box_off, box_raw, sc, ar, ob, os);
}